// ContrastiveLoss_9766755631545
// MI455X (gfx1250) — compile-verified
//
#include <hip/hip_runtime.h>
#include <hip/hip_bf16.h>

typedef __attribute__((ext_vector_type(2))) float v2f;
typedef __attribute__((ext_vector_type(8))) float v8f;

#define B_      16
#define T_      2048
#define F_      512
#define C_      256
#define NEG_    10
#define STEPS_  12
#define COPIES_ 11

// ---------------------------------------------------------------------------
// Phase 1: tl[b*T + t, c] = true_latent[b*T+t, :] @ Wl[:, c] + bl[c]
// M = B*T = 32768, K = 512, N = 256.  One block = 16 rows x 128 cols (8 waves,
// 16 cols each).  A-strip staged in LDS with +4-float row pad (bank-conflict
// free for the strided per-row ds_load_b64 of the A fragment).
// ---------------------------------------------------------------------------
__global__ __launch_bounds__(256) void proj_gemm_kernel(
    const float* __restrict__ A,   // [M, 512]
    const float* __restrict__ W,   // [512, 256]
    const float* __restrict__ bl,  // [256]
    float* __restrict__ tl) {      // [M, 256]
  __shared__ float sA[16 * 516];
  const int tid = threadIdx.x;
  const int m0 = blockIdx.x * 16;
  const int n0base = blockIdx.y * 128;

  // Cooperative load of the 16x512 f32 A strip (float4 vectorized).
  for (int idx = tid; idx < 16 * 128; idx += 256) {
    const int r = idx >> 7;       // row 0..15
    const int c4 = idx & 127;     // float4 index 0..127
    const float4 v = *(const float4*)(A + (size_t)(m0 + r) * F_ + c4 * 4);
    *(float4*)(&sA[r * 516 + c4 * 4]) = v;
  }
  __syncthreads();

  const int wave = tid >> 5;
  const int lane = tid & 31;
  const int hf = lane >> 4;       // half-wave: selects K pair
  const int rc = lane & 15;       // A-row / B-col within tile
  const int n0 = n0base + wave * 16;

  v8f acc = {};
  for (int k0 = 0; k0 < F_; k0 += 4) {
    const int k = k0 + 2 * hf;
    v2f a, b;
    a.x = sA[rc * 516 + k];
    a.y = sA[rc * 516 + k + 1];
    b.x = W[(size_t)k * C_ + n0 + rc];
    b.y = W[(size_t)(k + 1) * C_ + n0 + rc];
    acc = __builtin_amdgcn_wmma_f32_16x16x4_f32(false, a, false, b,
                                                (short)0, acc, false, false);
  }

  const float bias = bl[n0 + rc];
  for (int v = 0; v < 8; ++v) {
    tl[(size_t)(m0 + v + 8 * hf) * C_ + n0 + rc] = acc[v] + bias;
  }
}

// ---------------------------------------------------------------------------
// Phase 2: one wave per (b, t').  16x16 WMMA tile over K = 256:
//   A row i  (i<12, i<=t') = ctx[b, t'-i, :] * Ws[i] + bs[i]   (pred vector)
//   B col 0              = tl[b, t', :]                        (positive)
//   B col 1..10          = tl[b, neg_idx[b, c-1, t'], :]       (negatives)
// Result row i, cols 0..10 are the 11 logits of loss row (b, t'-i, i).
// Per row: logsumexp over cols 0..10 minus col 0; partials reduced per block.
// ---------------------------------------------------------------------------
__global__ __launch_bounds__(256) void logits_loss_kernel(
    const float* __restrict__ ctx,     // [B, T, 256]
    const float* __restrict__ tl,      // [B, T, 256]
    const float* __restrict__ Ws,      // [12]
    const float* __restrict__ bs,      // [12]
    const int* __restrict__ negidx,    // [B, 10, T]
    float* __restrict__ partial) {
  __shared__ float ssum[8];
  const int tid = threadIdx.x;
  const int wave = tid >> 5;
  const int lane = tid & 31;
  const int hf = lane >> 4;
  const int rc = lane & 15;
  const int b = blockIdx.y;
  const int tp = blockIdx.x * 8 + wave;   // t'

  // A-fragment row setup (row index rc = prediction step i)
  const bool rowValid = (rc < STEPS_) && (rc <= tp);
  const int srow = rowValid ? (tp - rc) : tp;
  const float wsv = rowValid ? Ws[rc] : 0.0f;
  const float bsv = rowValid ? bs[rc] : 0.0f;
  const float* aptr = ctx + ((size_t)b * T_ + srow) * C_;

  // B-fragment column setup (col index rc = copy c; 0 = positive)
  const bool colValid = (rc < COPIES_);
  int trow = tp;
  if (rc >= 1 && rc <= NEG_) {
    trow = negidx[((size_t)b * NEG_ + (rc - 1)) * T_ + tp];
  }
  const float colv = colValid ? 1.0f : 0.0f;
  const float* bptr = tl + ((size_t)b * T_ + trow) * C_;

  v8f acc = {};
  for (int k0 = 0; k0 < C_; k0 += 4) {
    const int k = k0 + 2 * hf;
    const float2 av = *(const float2*)(aptr + k);
    const float2 bv = *(const float2*)(bptr + k);
    v2f a, bb;
    a.x = av.x * wsv + bsv;
    a.y = av.y * wsv + bsv;
    bb.x = bv.x * colv;
    bb.y = bv.y * colv;
    acc = __builtin_amdgcn_wmma_f32_16x16x4_f32(false, a, false, bb,
                                                (short)0, acc, false, false);
  }

  // Per-row logsumexp - logit[col 0].  Row i lives in half-wave i/8, vgpr i%8;
  // butterfly masks 1..8 stay inside each 16-lane half.
  float lsum = 0.0f;
  const int maxRow = (tp < STEPS_ - 1) ? tp : (STEPS_ - 1);
  for (int v = 0; v < 8; ++v) {
    const int i = v + 8 * hf;
    const float x = acc[v];
    float mx = colValid ? x : -3.0e38f;
    for (int m = 1; m <= 8; m <<= 1) mx = fmaxf(mx, __shfl_xor(mx, m));
    float e = colValid ? __expf(x - mx) : 0.0f;
    for (int m = 1; m <= 8; m <<= 1) e += __shfl_xor(e, m);
    if (rc == 0 && i <= maxRow) {
      lsum += __logf(e) + mx - x;   // x here IS the col-0 logit of row i
    }
  }
  // Wave reduction (full 32-lane butterfly; only lanes 0 and 16 carry data).
  for (int m = 1; m <= 16; m <<= 1) lsum += __shfl_xor(lsum, m);
  if (lane == 0) ssum[wave] = lsum;
  __syncthreads();
  if (tid == 0) {
    float s = 0.0f;
    for (int w = 0; w < 8; ++w) s += ssum[w];
    partial[blockIdx.y * gridDim.x + blockIdx.x] = s;
  }
}

// ---------------------------------------------------------------------------
// Phase 3: deterministic tree reduction of block partials into d_out[0].
// ---------------------------------------------------------------------------
__global__ __launch_bounds__(256) void reduce_kernel(
    const float* __restrict__ partial, int n, float* __restrict__ out) {
  __shared__ float s[256];
  float acc = 0.0f;
  for (int i = threadIdx.x; i < n; i += 256) acc += partial[i];
  s[threadIdx.x] = acc;
  __syncthreads();
  for (int stride = 128; stride > 0; stride >>= 1) {
    if (threadIdx.x < stride) s[threadIdx.x] += s[threadIdx.x + stride];
    __syncthreads();
  }
  if (threadIdx.x == 0) out[0] = s[0];
}

extern "C" void kernel_launch(void* const* d_in, const int* in_sizes, int n_in,
                              void* d_out, int out_size, void* d_ws, size_t ws_size,
                              hipStream_t stream) {
  (void)in_sizes; (void)n_in; (void)out_size; (void)ws_size;
  const float* true_latent = (const float*)d_in[0];   // [16, 2048, 512]
  const float* ctx         = (const float*)d_in[1];   // [16, 2048, 256]
  const float* Wl          = (const float*)d_in[2];   // [512, 256]
  const float* bl          = (const float*)d_in[3];   // [256]
  const float* Ws          = (const float*)d_in[4];   // [12]
  const float* bs          = (const float*)d_in[5];   // [12]
  const int*   neg         = (const int*)d_in[6];     // [16, 10, 2048]

  float* tl      = (float*)d_ws;                      // 32 MiB: [B*T, 256]
  float* partial = tl + (size_t)B_ * T_ * C_;         // 4096 floats

  // Phase 1: projection GEMM (WMMA f32 16x16x4), grid 2048 x 2.
  proj_gemm_kernel<<<dim3((B_ * T_) / 16, C_ / 128), 256, 0, stream>>>(
      true_latent, Wl, bl, tl);

  // Phase 2: one 16x16 WMMA logit tile per (b, t'), 8 tiles per block.
  logits_loss_kernel<<<dim3(T_ / 8, B_), 256, 0, stream>>>(
      ctx, tl, Ws, bs, neg, partial);

  // Phase 3: final reduction.
  reduce_kernel<<<1, 256, 0, stream>>>(partial, (T_ / 8) * B_, (float*)d_out);
}